// GATV2_10917806867143
// MI455X (gfx1250) — compile-verified
//
#include <hip/hip_runtime.h>

#define N_NODES 50000
#define N_EDGES 400000
#define NEG_SLOPE 0.2f

typedef __attribute__((ext_vector_type(2))) float v2f;
typedef __attribute__((ext_vector_type(8))) float v8f;

// ---------------------------------------------------------------------------
// CSR build (dst-sorted edge lists, deterministic via per-node sort)
// ---------------------------------------------------------------------------
__global__ void hist_kernel(const int* __restrict__ dst, int* __restrict__ counts, int E) {
    int i = blockIdx.x * blockDim.x + threadIdx.x;
    if (i < E) atomicAdd(&counts[dst[i]], 1);
}

__global__ __launch_bounds__(1024)
void scan_kernel(const int* __restrict__ counts, int* __restrict__ offs, int n) {
    __shared__ int sh[1024];
    __shared__ int carry;
    if (threadIdx.x == 0) { carry = 0; offs[0] = 0; }
    __syncthreads();
    for (int base = 0; base < n; base += 1024) {
        int i = base + (int)threadIdx.x;
        sh[threadIdx.x] = (i < n) ? counts[i] : 0;
        __syncthreads();
        for (int o = 1; o < 1024; o <<= 1) {
            int t = ((int)threadIdx.x >= o) ? sh[threadIdx.x - o] : 0;
            __syncthreads();
            sh[threadIdx.x] += t;
            __syncthreads();
        }
        if (i < n) offs[i + 1] = sh[threadIdx.x] + carry;
        int total = sh[1023];
        __syncthreads();
        if (threadIdx.x == 0) carry += total;
        __syncthreads();
    }
}

__global__ void scatter_kernel(const int* __restrict__ dst, const int* __restrict__ offs,
                               int* __restrict__ cursor, int* __restrict__ perm, int E) {
    int i = blockIdx.x * blockDim.x + threadIdx.x;
    if (i < E) {
        int d = dst[i];
        int pos = offs[d] + atomicAdd(&cursor[d], 1);
        perm[pos] = i;
    }
}

__global__ void sortseg_kernel(const int* __restrict__ offs, int* __restrict__ perm, int n) {
    int v = blockIdx.x * blockDim.x + threadIdx.x;
    if (v >= n) return;
    int b = offs[v], e = offs[v + 1];
    for (int i = b + 1; i < e; ++i) {
        int key = perm[i];
        int j = i - 1;
        while (j >= b && perm[j] > key) { perm[j + 1] = perm[j]; --j; }
        perm[j + 1] = key;
    }
}

// ---------------------------------------------------------------------------
// FP32 GEMM, fast path: requires N % 64 == 0 and K % 32 == 0.
// C[M,N] = A[M,K] @ B[K,N] via V_WMMA_F32_16X16X4_F32.
// Wave tile 16(M) x 64(N); 8 waves/block -> 128 x 64 per block.
// B panel staged in LDS, pair-interleaved so a lane's B fragment
// (two K-adjacent values) is one ds_load_b64:
//   shB[(k>>1)*128 + 2*n + (k&1)] = B[kk+k][n0+n]
// A-frag: lane l -> row (l&15), K = 2*(l>>4)+{0,1}. C/D: row q+8*(l>>4), col l&15.
// ---------------------------------------------------------------------------
__global__ __launch_bounds__(256)
void gemm_wmma_full(const float* __restrict__ A, const float* __restrict__ B,
                    float* __restrict__ C, int M, int K, int N) {
    __shared__ float shB[32 * 64];
    const int tid   = threadIdx.x;
    const int lane  = tid & 31;
    const int wave  = tid >> 5;
    const int mtile = blockIdx.y * 8 + wave;
    const int n0    = blockIdx.x * 64;
    const int r     = lane & 15;
    const int hi    = lane >> 4;
    const bool mok  = (mtile * 16 < M);
    const int mrow  = (mok ? mtile : 0) * 16 + r;   // clamp so idle waves load safely

    const float* arow = A + (size_t)mrow * K + 2 * hi;

    // staging coordinates: 16 rows x 64 cols per pass, two passes per chunk
    const int skrow = tid >> 4;        // 0..15
    const int sncol = (tid & 15) * 4;  // 0..60

    v8f acc[4] = {};
    for (int kk = 0; kk < K; kk += 32) {
        __syncthreads();
#pragma unroll
        for (int it = 0; it < 2; ++it) {
            const int krow = it * 16 + skrow;
            const float4 vv = *(const float4*)(B + (size_t)(kk + krow) * N + n0 + sncol);
            float* sp = &shB[(krow >> 1) * 128 + sncol * 2 + (krow & 1)];
            sp[0] = vv.x; sp[2] = vv.y; sp[4] = vv.z; sp[6] = vv.w;
        }
        __syncthreads();
#pragma unroll
        for (int kl = 0; kl < 32; kl += 4) {
            v2f a;
            a.x = arow[kk + kl];
            a.y = arow[kk + kl + 1];
            const float* bp = &shB[((kl >> 1) + hi) * 128 + r * 2];
#pragma unroll
            for (int s = 0; s < 4; ++s) {
                const v2f b = *(const v2f*)(bp + s * 32);
                acc[s] = __builtin_amdgcn_wmma_f32_16x16x4_f32(
                    false, a, false, b, (short)0, acc[s], false, false);
            }
        }
    }
    if (!mok) return;
    const int rowbase = mtile * 16 + hi * 8;
#pragma unroll
    for (int s = 0; s < 4; ++s) {
        const int n = n0 + s * 16 + r;
#pragma unroll
        for (int q = 0; q < 8; ++q)
            C[(size_t)(rowbase + q) * N + n] = acc[s][q];
    }
}

// ---------------------------------------------------------------------------
// FP32 GEMM, edge path (N < 64, tiny layers): branchless column guard via
// clamped index + zero weight -> no EXEC manipulation in the K loop.
// ---------------------------------------------------------------------------
__global__ __launch_bounds__(256)
void gemm_wmma_edge(const float* __restrict__ A, const float* __restrict__ B,
                    float* __restrict__ C, int M, int K, int N) {
    const int lane  = threadIdx.x & 31;
    const int wave  = threadIdx.x >> 5;
    const int mtile = blockIdx.y * 8 + wave;
    if (mtile * 16 >= M) return;
    const int n0 = blockIdx.x * 64;
    const int r  = lane & 15;
    const int hi = lane >> 4;

    const float* arow = A + (size_t)(mtile * 16 + r) * K + 2 * hi;

    int   nc[4];
    float wt[4];
#pragma unroll
    for (int s = 0; s < 4; ++s) {
        const int n = n0 + s * 16 + r;
        nc[s] = (n < N) ? n : (N - 1);
        wt[s] = (n < N) ? 1.f : 0.f;
    }

    v8f acc[4] = {};
    for (int kk = 0; kk < K; kk += 4) {
        v2f a;
        a.x = arow[kk];
        a.y = arow[kk + 1];
        const float* b0 = B + (size_t)(kk + 2 * hi) * N;
        const float* b1 = b0 + N;
#pragma unroll
        for (int s = 0; s < 4; ++s) {
            v2f b;
            b.x = b0[nc[s]] * wt[s];
            b.y = b1[nc[s]] * wt[s];
            acc[s] = __builtin_amdgcn_wmma_f32_16x16x4_f32(
                false, a, false, b, (short)0, acc[s], false, false);
        }
    }
    const int rowbase = mtile * 16 + hi * 8;
#pragma unroll
    for (int s = 0; s < 4; ++s) {
        const int n = n0 + s * 16 + r;
        if (n >= N) continue;
#pragma unroll
        for (int q = 0; q < 8; ++q)
            C[(size_t)(rowbase + q) * N + n] = acc[s][q];
    }
}

// ---------------------------------------------------------------------------
// Edge scores: one wave per edge; lanes stripe the D dimension.
// ---------------------------------------------------------------------------
__global__ __launch_bounds__(256)
void edge_score_kernel(const float* __restrict__ fs, const float* __restrict__ fd,
                       const int* __restrict__ src, const int* __restrict__ dst,
                       const float* __restrict__ attn, float* __restrict__ score,
                       int H, int D) {
    const int lane = threadIdx.x & 31;
    const int e    = blockIdx.x * 8 + (threadIdx.x >> 5);
    if (e >= N_EDGES) return;
    const int HD = H * D;
    const size_t sb = (size_t)src[e] * HD;
    const size_t db = (size_t)dst[e] * HD;
    for (int h = 0; h < H; ++h) {
        float acc = 0.f;
        for (int j = lane; j < D; j += 32) {
            float v = fs[sb + h * D + j] + fd[db + h * D + j];
            v = (v > 0.f) ? v : NEG_SLOPE * v;
            acc += v * attn[h * D + j];
        }
#pragma unroll
        for (int o = 16; o > 0; o >>= 1) acc += __shfl_xor(acc, o, 32);
        if (lane == 0) score[(size_t)e * H + h] = acc;
    }
}

// ---------------------------------------------------------------------------
// Per-node edge softmax + weighted aggregation (no float atomics).
// ---------------------------------------------------------------------------
__global__ __launch_bounds__(256)
void node_agg_kernel(const float* __restrict__ fs, const float* __restrict__ score,
                     const int* __restrict__ src, const int* __restrict__ offs,
                     const int* __restrict__ perm, const float* __restrict__ bias,
                     float* __restrict__ out, int H, int D) {
    const int v  = blockIdx.x;
    const int HD = H * D;
    __shared__ float sm[4];
    __shared__ float siz[4];
    const int beg = offs[v], end = offs[v + 1];
    if ((int)threadIdx.x < H) {
        const int h = threadIdx.x;
        float m = -3.402823466e38f;
        for (int i = beg; i < end; ++i)
            m = fmaxf(m, score[(size_t)perm[i] * H + h]);
        float z = 0.f;
        for (int i = beg; i < end; ++i)
            z += __expf(score[(size_t)perm[i] * H + h] - m);
        sm[h]  = m;
        siz[h] = (z > 0.f) ? (1.f / z) : 1.f;
    }
    __syncthreads();
    for (int f = threadIdx.x; f < HD; f += 256) {
        const int h = f / D;
        const float m = sm[h], iz = siz[h];
        float acc = 0.f;
        for (int i = beg; i < end; ++i) {
            const int e = perm[i];
            const float c = __expf(score[(size_t)e * H + h] - m) * iz;
            acc += c * fs[(size_t)src[e] * HD + f];
        }
        out[(size_t)v * HD + f] = acc + bias[f];
    }
}

// ---------------------------------------------------------------------------
extern "C" void kernel_launch(void* const* d_in, const int* in_sizes, int n_in,
                              void* d_out, int out_size, void* d_ws, size_t ws_size,
                              hipStream_t stream) {
    (void)in_sizes; (void)n_in; (void)out_size; (void)ws_size;

    const float* x_in = (const float*)d_in[0];
    const int*   src  = (const int*)d_in[1];
    const int*   dst  = (const int*)d_in[2];

    char* ws = (char*)d_ws;
    size_t off = 0;
    auto alloc = [&](size_t bytes) -> void* {
        void* p = ws + off;
        off += (bytes + 255) & ~(size_t)255;
        return p;
    };
    float* big[3];
    for (int i = 0; i < 3; ++i)
        big[i] = (float*)alloc((size_t)N_NODES * 512 * sizeof(float));
    float* score  = (float*)alloc((size_t)N_EDGES * 4 * sizeof(float));
    int*   counts = (int*)alloc((size_t)N_NODES * sizeof(int));
    int*   offs   = (int*)alloc((size_t)(N_NODES + 1) * sizeof(int));
    int*   cursor = (int*)alloc((size_t)N_NODES * sizeof(int));
    int*   perm   = (int*)alloc((size_t)N_EDGES * sizeof(int));

    hipMemsetAsync(counts, 0, (size_t)N_NODES * sizeof(int), stream);
    hipMemsetAsync(cursor, 0, (size_t)N_NODES * sizeof(int), stream);
    hist_kernel<<<(N_EDGES + 255) / 256, 256, 0, stream>>>(dst, counts, N_EDGES);
    scan_kernel<<<1, 1024, 0, stream>>>(counts, offs, N_NODES);
    scatter_kernel<<<(N_EDGES + 255) / 256, 256, 0, stream>>>(dst, offs, cursor, perm, N_EDGES);
    sortseg_kernel<<<(N_NODES + 255) / 256, 256, 0, stream>>>(offs, perm, N_NODES);

    static const int DIN[6] = {256, 512, 256, 128, 64, 32};
    static const int DH[6]  = {128, 64, 32, 16, 8, 5};
    static const int NH[6]  = {4, 4, 4, 4, 4, 1};

    int xi = -1;  // -1 => layer input is d_in[0]
    for (int L = 0; L < 6; ++L) {
        const float* x = (xi < 0) ? x_in : big[xi];
        const int si = (xi < 0) ? 0 : (xi + 1) % 3;
        const int di = (xi < 0) ? 1 : (xi + 2) % 3;
        float* fS  = big[si];
        float* fD  = big[di];
        float* out = (L == 5) ? (float*)d_out : big[di];  // fD dead after scores

        const float* Wsrc = (const float*)d_in[3 + 4 * L + 0];
        const float* Wdst = (const float*)d_in[3 + 4 * L + 1];
        const float* attn = (const float*)d_in[3 + 4 * L + 2];
        const float* bias = (const float*)d_in[3 + 4 * L + 3];

        const int K = DIN[L], H = NH[L], D = DH[L], HD = H * D;

        dim3 gg((HD + 63) / 64, (N_NODES / 16 + 7) / 8);
        if (HD % 64 == 0 && K % 32 == 0) {
            gemm_wmma_full<<<gg, 256, 0, stream>>>(x, Wsrc, fS, N_NODES, K, HD);
            gemm_wmma_full<<<gg, 256, 0, stream>>>(x, Wdst, fD, N_NODES, K, HD);
        } else {
            gemm_wmma_edge<<<gg, 256, 0, stream>>>(x, Wsrc, fS, N_NODES, K, HD);
            gemm_wmma_edge<<<gg, 256, 0, stream>>>(x, Wdst, fD, N_NODES, K, HD);
        }

        edge_score_kernel<<<(N_EDGES + 7) / 8, 256, 0, stream>>>(
            fS, fD, src, dst, attn, score, H, D);

        node_agg_kernel<<<N_NODES, 256, 0, stream>>>(
            fS, score, src, offs, perm, bias, out, H, D);

        xi = di;
    }
}